// GraphConvolution_10900626998074
// MI455X (gfx1250) — compile-verified
//
#include <hip/hip_runtime.h>
#include <hip/hip_bf16.h>

#define N_NODES 100000
#define N_EDGES 1000000
#define D 64

typedef __attribute__((ext_vector_type(2))) float v2f;
typedef __attribute__((ext_vector_type(8))) float v8f;

// ---------------------------------------------------------------------------
// Zero out[N_NODES*D] and deg[N_NODES] (d_out / d_ws are poisoned 0xAA).
// ---------------------------------------------------------------------------
__global__ void gcn_zero_kernel(float* __restrict__ out, float* __restrict__ deg) {
    int i = blockIdx.x * blockDim.x + threadIdx.x;
    const int nOut = N_NODES * D;
    if (i < nOut) {
        out[i] = 0.0f;
    } else if (i < nOut + N_NODES) {
        deg[i - nOut] = 0.0f;
    }
}

// ---------------------------------------------------------------------------
// Y = X @ W  via V_WMMA_F32_16X16X4_F32.
// Block = 256 threads = 8 waves. Each block handles 32 rows of X:
//   2 row-subtiles (16 rows) x 4 col-tiles (16 cols) = 8 wave-tiles.
// W (64x64, 16KB) and the 32x64 X slab (8KB) staged in LDS.
// ---------------------------------------------------------------------------
__global__ __launch_bounds__(256) void gcn_gemm_wmma_kernel(
    const float* __restrict__ x, const float* __restrict__ w,
    float* __restrict__ y)
{
    __shared__ float sW[64 * 64];        // full weight matrix
    __shared__ float sX[32 * 64];        // 2 row-subtiles of X

    const int tid = threadIdx.x;
    const int rowBase = blockIdx.x * 32; // 3125 blocks * 32 = 100000 exactly

    // Cooperative staging, float4-wide (coalesced).
    {
        const float4* src = (const float4*)w;
        float4* dst = (float4*)sW;
        #pragma unroll
        for (int i = tid; i < 64 * 64 / 4; i += 256) dst[i] = src[i];
    }
    {
        const float4* src = (const float4*)(x + (size_t)rowBase * D);
        float4* dst = (float4*)sX;
        #pragma unroll
        for (int i = tid; i < 32 * 64 / 4; i += 256) dst[i] = src[i];
    }
    __syncthreads();

    const int wave    = tid >> 5;         // 0..7
    const int lane    = tid & 31;
    const int subTile = wave >> 2;        // 0..1 (which 16-row subtile)
    const int n0      = (wave & 3) * 16;  // output column tile
    const int m       = lane & 15;
    const int kb      = (lane < 16) ? 0 : 2;  // K-pair base for this half-wave

    const float* ax = sX + (subTile * 16 + m) * D; // this lane's X row

    v8f c = {};
    #pragma unroll
    for (int s = 0; s < 16; ++s) {
        const int k = 4 * s;
        v2f a, b;
        // A 16x4 layout: VGPR0=K kb+0, VGPR1=K kb+1 (lanes 0-15: kb=0; 16-31: kb=2)
        a.x = ax[k + kb + 0];
        a.y = ax[k + kb + 1];
        // B 4x16 layout: VGPR0 = row K=kb+0, VGPR1 = row K=kb+1, N = m
        b.x = sW[(k + kb + 0) * D + n0 + m];
        b.y = sW[(k + kb + 1) * D + n0 + m];
        c = __builtin_amdgcn_wmma_f32_16x16x4_f32(
                /*neg_a=*/false, a, /*neg_b=*/false, b,
                /*c_mod=*/(short)0, c, /*reuse_a=*/false, /*reuse_b=*/false);
    }

    // C/D layout: VGPR i -> M = i + (lane<16 ? 0 : 8), N = m
    const int mRow = rowBase + subTile * 16 + ((lane < 16) ? 0 : 8);
    #pragma unroll
    for (int i = 0; i < 8; ++i) {
        y[(size_t)(mRow + i) * D + n0 + m] = c[i];
    }
}

// ---------------------------------------------------------------------------
// deg[row] += 1 for each edge (f32 atomics, resolved in L2).
// ---------------------------------------------------------------------------
__global__ void gcn_degree_kernel(const long long* __restrict__ rows,
                                  float* __restrict__ deg) {
    int e = blockIdx.x * blockDim.x + threadIdx.x;
    if (e < N_EDGES) {
        atomicAdd(&deg[(int)rows[e]], 1.0f);
    }
}

// ---------------------------------------------------------------------------
// deg[i] = rsqrt(max(deg[i], 1))   (in place -> deg_inv_sqrt)
// ---------------------------------------------------------------------------
__global__ void gcn_rsqrt_kernel(float* __restrict__ deg) {
    int i = blockIdx.x * blockDim.x + threadIdx.x;
    if (i < N_NODES) {
        float d = deg[i];
        d = (d < 1.0f) ? 1.0f : d;
        deg[i] = rsqrtf(d);
    }
}

// ---------------------------------------------------------------------------
// out[row] += Y[col] * (dis[row]*dis[col]).
// 16 threads per edge, each owning one float4 chunk of the 64-dim feature.
// 128-bit gathers; 4 scalar f32 atomics per thread into out (L2-resident).
// ---------------------------------------------------------------------------
__global__ __launch_bounds__(256) void gcn_scatter_kernel(
    const float* __restrict__ y,
    const long long* __restrict__ rows,
    const long long* __restrict__ cols,
    const float* __restrict__ dis,
    float* __restrict__ out)
{
    const long long t = (long long)blockIdx.x * blockDim.x + threadIdx.x;
    const int e = (int)(t >> 4);
    if (e >= N_EDGES) return;
    const int c4 = (int)(t & 15) * 4;

    const int r  = (int)rows[e];
    const int cl = (int)cols[e];
    const float norm = dis[r] * dis[cl];

    const float4 v = *(const float4*)(y + (size_t)cl * D + c4);
    float* o = out + (size_t)r * D + c4;
    atomicAdd(o + 0, v.x * norm);
    atomicAdd(o + 1, v.y * norm);
    atomicAdd(o + 2, v.z * norm);
    atomicAdd(o + 3, v.w * norm);
}

// ---------------------------------------------------------------------------
extern "C" void kernel_launch(void* const* d_in, const int* in_sizes, int n_in,
                              void* d_out, int out_size, void* d_ws, size_t ws_size,
                              hipStream_t stream) {
    const float*     x    = (const float*)d_in[0];      // [N_NODES, 64]
    const long long* edge = (const long long*)d_in[1];  // [2, N_EDGES] int64
    const float*     w    = (const float*)d_in[2];      // [64, 64]
    float* out = (float*)d_out;                         // [N_NODES, 64]

    // Workspace layout: deg/dis at 0 (400KB), Y at 1MB offset (25.6MB).
    float* deg = (float*)d_ws;
    float* y   = (float*)((char*)d_ws + (1u << 20));

    const long long* rows = edge;
    const long long* cols = edge + N_EDGES;

    {   // zero out + deg
        const int total = N_NODES * D + N_NODES;
        gcn_zero_kernel<<<(total + 255) / 256, 256, 0, stream>>>(out, deg);
    }
    // Y = X @ W  (WMMA fp32)
    gcn_gemm_wmma_kernel<<<(N_NODES / 32), 256, 0, stream>>>(x, w, y);
    // degree accumulation
    gcn_degree_kernel<<<(N_EDGES + 255) / 256, 256, 0, stream>>>(rows, deg);
    // deg -> deg^{-1/2}
    gcn_rsqrt_kernel<<<(N_NODES + 255) / 256, 256, 0, stream>>>(deg);
    // fused gather * norm -> scatter-add
    {
        const long long nThreads = (long long)N_EDGES * 16;
        const int nBlocks = (int)((nThreads + 255) / 256);
        gcn_scatter_kernel<<<nBlocks, 256, 0, stream>>>(y, rows, cols, deg, out);
    }
}